// rec_attention_78469052498198
// MI455X (gfx1250) — compile-verified
//
#include <hip/hip_runtime.h>
#include <hip/hip_bf16.h>
#include <math.h>

typedef __attribute__((ext_vector_type(2))) float v2f;
typedef __attribute__((ext_vector_type(8))) float v8f;

namespace {
constexpr int kB = 1024;
constexpr int kS = 2048;
constexpr int kD = 128;
constexpr int kWaves = 8;
constexpr int kThreads = kWaves * 32;          // 256 threads = 8 wave32
constexpr int kRowsPerWave = kS / kWaves;      // 256 rows per wave
constexpr int kTilesPerWave = kRowsPerWave / 16; // 16 tiles of 16 rows
}

// One workgroup per batch. Single HBM pass over X[b] (1 MB):
//   per 16-row tile: scores via V_WMMA_F32_16X16X4_F32 (fp32, k-stepped by 4),
//   then online-softmax update of per-wave (m, l, acc[128]) with the tile
//   re-read out of WGP$/L2. Raw scores go to LDS so alpha is written at the
//   end without re-touching HBM. Waves combine through LDS at the end.
__launch_bounds__(kThreads, 1)
__global__ void attn_pool_wmma(const float* __restrict__ X,
                               const float* __restrict__ V,
                               float* __restrict__ out_repres,
                               float* __restrict__ out_alpha)
{
  __shared__ __align__(16) float s_scores[kS];        // 8 KB raw scores
  __shared__ __align__(16) float s_v[kD];             // context vector
  __shared__ float s_m[kWaves];
  __shared__ float s_l[kWaves];
  __shared__ __align__(16) float s_acc[kWaves][kD];   // per-wave accumulators

  const int bidx = blockIdx.x;
  const int tid  = threadIdx.x;
  const int wave = tid >> 5;
  const int lane = tid & 31;

  const float* __restrict__ Xb = X + (size_t)bidx * kS * kD;

  if (tid < kD) s_v[tid] = V[tid];
  __syncthreads();

  // WMMA 16x16x4 f32 fragment addressing for this lane:
  //   A: row M = lane&15, VGPR pair holds K = koff, koff+1 with koff = 2*(lane>=16)
  //   B: same K striping; we replicate v across all 16 N columns.
  const int koff = (lane >> 4) << 1;   // 0 or 2
  const int mrow = lane & 15;

  // Hoist loop-invariant B fragments (v) into registers: 32 k-steps of 4.
  v2f bfrag[kD / 4];
#pragma unroll
  for (int k = 0; k < kD / 4; ++k)
    bfrag[k] = *(const v2f*)&s_v[k * 4 + koff];

  float m_run = -INFINITY;
  float l_run = 0.0f;
  float acc0 = 0.f, acc1 = 0.f, acc2 = 0.f, acc3 = 0.f;

  const int row_base = wave * kRowsPerWave;

#pragma unroll 1
  for (int t = 0; t < kTilesPerWave; ++t) {
    const int r0 = row_base + t * 16;

    // Hint the next tile into cache while this one computes.
    if (t + 1 < kTilesPerWave) {
      __builtin_prefetch(&Xb[(size_t)(r0 + 16 + mrow) * kD + (koff << 5)], 0, 0);
    }

    // ---- scores for 16 rows via 32 chained f32 WMMAs (K stepped by 4) ----
    const float* __restrict__ arow = &Xb[(size_t)(r0 + mrow) * kD];
    v8f c = {};
#pragma unroll
    for (int k0 = 0; k0 < kD; k0 += 4) {
      const v2f a = *(const v2f*)(arow + k0 + koff);   // global_load_b64
      c = __builtin_amdgcn_wmma_f32_16x16x4_f32(
          /*neg_a=*/false, a, /*neg_b=*/false, bfrag[k0 >> 2],
          /*c_mod=*/(short)0, c, /*reuse_a=*/false, /*reuse_b=*/false);
    }

    // C layout: VGPR r holds M=r (lanes 0-15) / M=8+r (lanes 16-31); every N
    // column is identical since B was replicated. Broadcast all 16 scores.
    float sc[16];
#pragma unroll
    for (int r = 0; r < 8; ++r) {
      sc[r]     = __shfl(c[r], 0, 32);
      sc[r + 8] = __shfl(c[r], 16, 32);
    }

    // Stash raw scores for the final alpha write (two b128 DS stores).
    if (lane == 0 || lane == 16) {
      const int off = r0 + ((lane >> 4) << 3);
      *(float4*)&s_scores[off]     = make_float4(c[0], c[1], c[2], c[3]);
      *(float4*)&s_scores[off + 4] = make_float4(c[4], c[5], c[6], c[7]);
    }

    // ---- online softmax update (tile-granular rescale) ----
    float tmax = sc[0];
#pragma unroll
    for (int r = 1; r < 16; ++r) tmax = fmaxf(tmax, sc[r]);
    const float m_new = fmaxf(m_run, tmax);
    const float rescale = __expf(m_run - m_new);   // exp(-inf)=0 on first tile
    l_run *= rescale;
    acc0 *= rescale; acc1 *= rescale; acc2 *= rescale; acc3 *= rescale;

#pragma unroll
    for (int r = 0; r < 16; ++r) {
      const float p = __expf(sc[r] - m_new);
      l_run += p;
      // Re-read the row lanes-over-D (coalesced 512B/row); hot in WGP$/L2.
      const float4 xv = *(const float4*)&Xb[(size_t)(r0 + r) * kD + lane * 4];
      acc0 = fmaf(p, xv.x, acc0);
      acc1 = fmaf(p, xv.y, acc1);
      acc2 = fmaf(p, xv.z, acc2);
      acc3 = fmaf(p, xv.w, acc3);
    }
    m_run = m_new;
  }

  // ---- cross-wave combine ----
  if (lane == 0) { s_m[wave] = m_run; s_l[wave] = l_run; }
  *(float4*)&s_acc[wave][lane * 4] = make_float4(acc0, acc1, acc2, acc3);
  __syncthreads();

  float M = s_m[0];
#pragma unroll
  for (int w = 1; w < kWaves; ++w) M = fmaxf(M, s_m[w]);
  float L = 0.f;
#pragma unroll
  for (int w = 0; w < kWaves; ++w) L += s_l[w] * __expf(s_m[w] - M);
  const float invL = 1.0f / L;

  if (tid < kD) {
    float r = 0.f;
#pragma unroll
    for (int w = 0; w < kWaves; ++w) r += s_acc[w][tid] * __expf(s_m[w] - M);
    out_repres[(size_t)bidx * kD + tid] = r * invL;
  }

#pragma unroll
  for (int s = 0; s < kS; s += kThreads) {
    out_alpha[(size_t)bidx * kS + s + tid] =
        __expf(s_scores[s + tid] - M) * invL;
  }
}

extern "C" void kernel_launch(void* const* d_in, const int* in_sizes, int n_in,
                              void* d_out, int out_size, void* d_ws, size_t ws_size,
                              hipStream_t stream) {
  (void)in_sizes; (void)n_in; (void)out_size; (void)d_ws; (void)ws_size;
  const float* X = (const float*)d_in[0];          // [B, S, D] fp32
  const float* V = (const float*)d_in[1];          // [D, 1]   fp32
  float* out_repres = (float*)d_out;               // first B*D floats
  float* out_alpha  = (float*)d_out + (size_t)kB * kD; // then B*S floats
  attn_pool_wmma<<<dim3(kB), dim3(kThreads), 0, stream>>>(X, V, out_repres, out_alpha);
}